// _MaxPool_11991548690484
// MI455X (gfx1250) — compile-verified
//
#include <hip/hip_runtime.h>

// Fused 2x2x2 spatial max-pool (+argmax idx) and k=2 temporal max-pool (+argmax idx)
// x: [B=2, C=8, T=8, D=64, H=128, W=128] fp32
// out (concat, in return order):
//   x_pool        [2,8,4,32,64,64]  float  (N1 = 8,388,608)
//   inds_spatial  [2,8,8,32,64,64]  int32  (N2 = 16,777,216)
//   inds_temporal [2,8,4,131072]    int32  (N3 = 8,388,608)
//
// Memory-bound: 512 MiB read + 128 MiB written, ~27us floor at 23.3 TB/s.
// One thread -> two adjacent wo outputs; 8x b128 NT loads per thread; a wave's
// 32 lanes * 16B = 512B = exactly one full input row (W=128 fp32) per load.

typedef float f4 __attribute__((ext_vector_type(4)));
typedef float f2 __attribute__((ext_vector_type(2)));
typedef int   i2 __attribute__((ext_vector_type(2)));

__device__ __forceinline__ void win8_argmax(f4 a, f4 b, f4 c, f4 d, int j,
                                            float& best_v, int& best_a) {
  // Window scan order (kd,kh,kw) lexicographic; strict > => first-max wins
  // (matches jnp.argmax / torch tie-breaking).
  // a = (kd0,kh0), b = (kd0,kh1), c = (kd1,kh0), d = (kd1,kh1)
  // j==0 -> components .x(.kw0)/.y(.kw1); j==1 -> .z/.w
  const float e0 = j ? a.z : a.x, e1 = j ? a.w : a.y;
  const float e2 = j ? b.z : b.x, e3 = j ? b.w : b.y;
  const float e4 = j ? c.z : c.x, e5 = j ? c.w : c.y;
  const float e6 = j ? d.z : d.x, e7 = j ? d.w : d.y;
  float v = e0; int g = 0;
  if (e1 > v) { v = e1; g = 1; }
  if (e2 > v) { v = e2; g = 2; }
  if (e3 > v) { v = e3; g = 3; }
  if (e4 > v) { v = e4; g = 4; }
  if (e5 > v) { v = e5; g = 5; }
  if (e6 > v) { v = e6; g = 6; }
  if (e7 > v) { v = e7; g = 7; }
  best_v = v; best_a = g;
}

__global__ __launch_bounds__(256) void maxpool3d_t_fused(
    const float* __restrict__ x,
    float* __restrict__ xpool,
    int*   __restrict__ inds_sp,
    int*   __restrict__ inds_t,
    long nthreads) {
  // Shape constants (reference is fixed-shape).
  constexpr int  W = 128, H = 128;
  constexpr int  T = 8, C = 8;
  constexpr int  Wo = 64, Ho = 64, Do = 32, To = 4;
  constexpr int  HALF = Wo / 2;                 // 32: wo-pairs per output row
  constexpr long PLANE = (long)H * W;           // 16384 floats (one D slice)
  constexpr long SLICE = (long)64 * PLANE;      // 1,048,576 floats per (b,c,t)
  constexpr long LSP   = (long)Do * Ho * Wo;    // 131072 per (b,c,t) output slice

  const long tid = (long)blockIdx.x * blockDim.x + threadIdx.x;
  if (tid >= nthreads) return;

  // tid = ((((bc)*To + tp)*Do + d0)*Ho + ho)*HALF + wq
  const int  wq = (int)(tid & (HALF - 1));
  long r  = tid >> 5;
  const int  ho = (int)(r & (Ho - 1)); r >>= 6;
  const int  d0 = (int)(r & (Do - 1)); r >>= 5;
  const int  tp = (int)(r & (To - 1));
  const long bc = r >> 2;                       // b*C + c, 0..15

  // Input base for (bc, t=2*tp, d=2*d0, h=2*ho, w=4*wq)
  const float* base = x + (bc * T + 2 * tp) * SLICE
                        + (long)(2 * d0) * PLANE
                        + (long)(2 * ho) * W
                        + 4 * wq;

  // 8 x 16B non-temporal loads (single-touch stream; don't pollute L2).
  f4 rv[2][4];
#pragma unroll
  for (int kt = 0; kt < 2; ++kt) {
#pragma unroll
    for (int kd = 0; kd < 2; ++kd) {
#pragma unroll
      for (int kh = 0; kh < 2; ++kh) {
        const float* p = base + (long)kt * SLICE + (long)kd * PLANE + kh * W;
        rv[kt][kd * 2 + kh] = __builtin_nontemporal_load((const f4*)p);
      }
    }
  }

  // Spatial pool + argmax for both t-slices and both wo columns.
  float vals[2][2];
  int   sidx[2][2];
#pragma unroll
  for (int kt = 0; kt < 2; ++kt) {
#pragma unroll
    for (int j = 0; j < 2; ++j) {
      float bv; int ba;
      win8_argmax(rv[kt][0], rv[kt][1], rv[kt][2], rv[kt][3], j, bv, ba);
      const int kd = ba >> 2, kh = (ba >> 1) & 1, kw = ba & 1;
      const int wo = 2 * wq + j;
      vals[kt][j] = bv;
      sidx[kt][j] = ((2 * d0 + kd) * H + (2 * ho + kh)) * W + (2 * wo + kw);
    }
  }

  // inds_spatial: [bc, t, d0, ho, wo]; inner offset within a (bc,t) slice is
  // 2*(tid mod 65536) because (d0,ho,wq) occupy the low 16 bits of tid.
  const long inner = (long)((tid & 0xFFFF) << 1);
#pragma unroll
  for (int kt = 0; kt < 2; ++kt) {
    i2 si; si.x = sidx[kt][0]; si.y = sidx[kt][1];
    int* p = inds_sp + (bc * T + 2 * tp + kt) * LSP + inner;
    __builtin_nontemporal_store(si, (i2*)p);
  }

  // Temporal pool: first-max tie-break => kt=1 only on strict >.
  const int k0 = vals[1][0] > vals[0][0];
  const int k1 = vals[1][1] > vals[0][1];
  f2 ov; ov.x = k0 ? vals[1][0] : vals[0][0];
         ov.y = k1 ? vals[1][1] : vals[0][1];
  i2 ot; ot.x = 2 * tp + k0;
         ot.y = 2 * tp + k1;

  // x_pool / inds_temporal share the same flat layout; element index = 2*tid+j.
  __builtin_nontemporal_store(ov, (f2*)(xpool  + 2 * tid));
  __builtin_nontemporal_store(ot, (i2*)(inds_t + 2 * tid));
}

extern "C" void kernel_launch(void* const* d_in, const int* in_sizes, int n_in,
                              void* d_out, int out_size, void* d_ws, size_t ws_size,
                              hipStream_t stream) {
  (void)in_sizes; (void)n_in; (void)out_size; (void)d_ws; (void)ws_size;

  const float* x = (const float*)d_in[0];

  constexpr long N1 = 8388608L;    // x_pool floats
  constexpr long N2 = 16777216L;   // inds_spatial int32
  float* out     = (float*)d_out;
  float* xpool   = out;
  int*   inds_sp = (int*)(out + N1);
  int*   inds_t  = (int*)(out + N1 + N2);

  const long nthreads = N1 / 2;    // 4,194,304 (two wo outputs per thread)
  const int  block = 256;          // 8 waves (wave32)
  const long grid  = (nthreads + block - 1) / block;   // 16384 blocks

  maxpool3d_t_fused<<<dim3((unsigned)grid), dim3(block), 0, stream>>>(
      x, xpool, inds_sp, inds_t, nthreads);
}